// RGCN_13864154432004
// MI455X (gfx1250) — compile-verified
//
#include <hip/hip_runtime.h>
#include <hip/hip_bf16.h>

// ---------------------------------------------------------------------------
// RGCN forward for MI455X (gfx1250).
//   layer: agg[dst] += (x[src] @ W[et]) * inv_cnt[dst,et];  h = relu(agg + x@root + b)
//   then global mean pool over sorted batch ids, then [64 -> 2] linear.
// Dense transforms use V_WMMA_F32_16X16X4_F32 (exact fp32). Edge phase is the
// bandwidth-bound part and uses coalesced gathers + global f32 atomics.
// ---------------------------------------------------------------------------

#define N_NODES_C  50000
#define N_REL_C    8
#define DIM_C      64
#define N_GRAPHS_C 64

typedef float v2f __attribute__((ext_vector_type(2)));
typedef float v8f __attribute__((ext_vector_type(8)));

// ---------------------------------------------------------------------------
// GEMM: for r in [0,8): XR[n, r, :] = X[n,:] @ W[r]   (W[r] is 64x64 row-major)
//       for r == 8:     XROOT[n, :] = X[n,:] @ root
// grid = (nNodes/16, 9), block = 128 (4 waves). Wave w computes cols [16w,16w+16).
// ---------------------------------------------------------------------------
__global__ void __launch_bounds__(128)
rgcn_wmma_gemm(const float* __restrict__ X, const float* __restrict__ W,
               const float* __restrict__ root,
               float* __restrict__ XR, float* __restrict__ XROOT, int nNodes) {
    __shared__ float xt[16][DIM_C + 1];   // padded: strided A reads avoid bank conflicts

    const int m0 = blockIdx.x * 16;
    const int r  = blockIdx.y;            // 0..7 = relations, 8 = root
    const float* __restrict__ Wr = (r < N_REL_C) ? (W + (size_t)r * DIM_C * DIM_C) : root;

    // Cooperative, fully coalesced load of the 16x64 X tile (1024 floats / 128 thr).
    const float* __restrict__ Xt = X + (size_t)m0 * DIM_C;
    for (int i = threadIdx.x; i < 16 * DIM_C; i += 128) {
        xt[i >> 6][i & 63] = Xt[i];
    }
    __syncthreads();

    const int lane = threadIdx.x & 31;
    const int wv   = threadIdx.x >> 5;    // 0..3 -> n0 = 16*wv
    const int n0   = wv * 16;
    const int row  = lane & 15;           // A-matrix M within tile
    const int col  = lane & 15;           // B/D N within tile
    const int kb   = (lane >> 4) << 1;    // lanes 0-15: K+{0,1}; lanes 16-31: K+{2,3}

    v8f acc = {};
    #pragma unroll
    for (int k = 0; k < DIM_C; k += 4) {
        v2f a, b;
        a.x = xt[row][k + kb];
        a.y = xt[row][k + kb + 1];
        b.x = Wr[(size_t)(k + kb)     * DIM_C + n0 + col];
        b.y = Wr[(size_t)(k + kb + 1) * DIM_C + n0 + col];
        // D = A(16x4,f32) * B(4x16,f32) + C ; 8-arg form:
        // (neg_a, A, neg_b, B, c_mod, C, reuse_a, reuse_b)
        acc = __builtin_amdgcn_wmma_f32_16x16x4_f32(false, a, false, b,
                                                    (short)0, acc, false, false);
    }

    // D layout: VGPR v -> (M = v + 8*(lane>=16), N = lane&15)
    const int mhi = (lane >> 4) << 3;
    if (r < N_REL_C) {
        #pragma unroll
        for (int v = 0; v < 8; ++v) {
            int m = m0 + v + mhi;
            XR[((size_t)m * N_REL_C + r) * DIM_C + n0 + col] = acc[v];
        }
    } else {
        #pragma unroll
        for (int v = 0; v < 8; ++v) {
            int m = m0 + v + mhi;
            XROOT[(size_t)m * DIM_C + n0 + col] = acc[v];
        }
    }
}

// ---------------------------------------------------------------------------
// Per-(dst, rel) incoming-edge counts (into INV buffer as float counts).
// ---------------------------------------------------------------------------
__global__ void rgcn_count_edges(const int* __restrict__ dst, const int* __restrict__ et,
                                 float* __restrict__ CNT, int nEdges) {
    int e = blockIdx.x * blockDim.x + threadIdx.x;
    if (e < nEdges) {
        atomicAdd(&CNT[(size_t)dst[e] * N_REL_C + et[e]], 1.0f);
    }
}

__global__ void rgcn_invert_counts(float* __restrict__ CNT, int n) {
    int i = blockIdx.x * blockDim.x + threadIdx.x;
    if (i < n) {
        float c = CNT[i];
        CNT[i] = 1.0f / fmaxf(c, 1.0f);
    }
}

// ---------------------------------------------------------------------------
// Edge aggregation: AGG[dst] += XR[src, et, :] * INV[dst, et]
// 16 threads per edge, 4 consecutive floats each (coalesced 256B gather).
// ---------------------------------------------------------------------------
__global__ void rgcn_edge_agg(const float* __restrict__ XR, const float* __restrict__ INV,
                              const int* __restrict__ src, const int* __restrict__ dst,
                              const int* __restrict__ et,
                              float* __restrict__ AGG, int nEdges) {
    long long gid = (long long)blockIdx.x * blockDim.x + threadIdx.x;
    int e  = (int)(gid >> 4);
    int c4 = ((int)gid & 15) << 2;
    if (e >= nEdges) return;
    int s = src[e], d = dst[e], r = et[e];
    float sc = INV[(size_t)d * N_REL_C + r];
    const float* __restrict__ m = XR + ((size_t)s * N_REL_C + r) * DIM_C + c4;
    float* __restrict__ o = AGG + (size_t)d * DIM_C + c4;
    float4 v = *(const float4*)m;
    atomicAdd(o + 0, v.x * sc);
    atomicAdd(o + 1, v.y * sc);
    atomicAdd(o + 2, v.z * sc);
    atomicAdd(o + 3, v.w * sc);
}

// ---------------------------------------------------------------------------
// H = relu(AGG + XROOT + b)
// ---------------------------------------------------------------------------
__global__ void rgcn_combine_relu(const float* __restrict__ AGG, const float* __restrict__ XROOT,
                                  const float* __restrict__ b, float* __restrict__ H, int nElems) {
    int i = blockIdx.x * blockDim.x + threadIdx.x;
    if (i < nElems) {
        float v = AGG[i] + XROOT[i] + b[i & (DIM_C - 1)];
        H[i] = fmaxf(v, 0.0f);
    }
}

// ---------------------------------------------------------------------------
// Global mean pool: PSUM[g, :] += H[n, :]; PCNT[g] += 1
// ---------------------------------------------------------------------------
__global__ void rgcn_pool(const float* __restrict__ H, const int* __restrict__ batch,
                          float* __restrict__ PSUM, float* __restrict__ PCNT, int nElems) {
    int i = blockIdx.x * blockDim.x + threadIdx.x;
    if (i < nElems) {
        int node = i >> 6;
        int c    = i & (DIM_C - 1);
        int g    = batch[node];
        atomicAdd(&PSUM[(size_t)g * DIM_C + c], H[i]);
        if (c == 0) atomicAdd(&PCNT[g], 1.0f);
    }
}

// ---------------------------------------------------------------------------
// out[g, c] = (PSUM[g,:]/max(PCNT[g],1)) @ lin_w[:,c] + lin_b[c]   (64x2)
// ---------------------------------------------------------------------------
__global__ void rgcn_final_linear(const float* __restrict__ PSUM, const float* __restrict__ PCNT,
                                  const float* __restrict__ lw, const float* __restrict__ lb,
                                  float* __restrict__ out) {
    int t = threadIdx.x;           // 128 threads: 64 graphs x 2 classes
    int g = t >> 1;
    int c = t & 1;
    float inv = 1.0f / fmaxf(PCNT[g], 1.0f);
    float acc = lb[c];
    #pragma unroll
    for (int h = 0; h < DIM_C; ++h) {
        acc += PSUM[(size_t)g * DIM_C + h] * inv * lw[h * 2 + c];
    }
    out[g * 2 + c] = acc;
}

// ---------------------------------------------------------------------------
// Host-side orchestration (graph-capture safe: only kernel launches and
// hipMemsetAsync on `stream`).
// ---------------------------------------------------------------------------
extern "C" void kernel_launch(void* const* d_in, const int* in_sizes, int n_in,
                              void* d_out, int out_size, void* d_ws, size_t ws_size,
                              hipStream_t stream) {
    const float* x     = (const float*)d_in[0];
    const float* w1    = (const float*)d_in[1];
    const float* root1 = (const float*)d_in[2];
    const float* b1    = (const float*)d_in[3];
    const float* w2    = (const float*)d_in[4];
    const float* root2 = (const float*)d_in[5];
    const float* b2    = (const float*)d_in[6];
    const float* lin_w = (const float*)d_in[7];
    const float* lin_b = (const float*)d_in[8];
    const int*   src   = (const int*)d_in[9];
    const int*   dst   = (const int*)d_in[10];
    const int*   et    = (const int*)d_in[11];
    const int*   batch = (const int*)d_in[12];
    float* out = (float*)d_out;

    const int nNodes = in_sizes[0] / DIM_C;     // 50000
    const int nEdges = in_sizes[9];             // 800000

    // Workspace layout (floats)
    char* base = (char*)d_ws;
    const size_t xrBytes    = (size_t)nNodes * N_REL_C * DIM_C * sizeof(float); // 102.4 MB
    const size_t nodeBytes  = (size_t)nNodes * DIM_C * sizeof(float);           // 12.8 MB
    const size_t invBytes   = (size_t)nNodes * N_REL_C * sizeof(float);         // 1.6 MB
    float* XR    = (float*)(base);
    float* XROOT = (float*)(base + xrBytes);
    float* AGG   = (float*)(base + xrBytes + nodeBytes);
    float* H1    = (float*)(base + xrBytes + 2 * nodeBytes);
    float* H2    = (float*)(base + xrBytes + 3 * nodeBytes);
    float* INV   = (float*)(base + xrBytes + 4 * nodeBytes);
    float* PSUM  = (float*)(base + xrBytes + 4 * nodeBytes + invBytes);
    float* PCNT  = PSUM + N_GRAPHS_C * DIM_C;

    const int nTilesM   = nNodes / 16;              // 3125 (exact)
    const int nodeElems = nNodes * DIM_C;

    // --- Per-(dst,rel) inverse mean counts (shared by both layers) ---
    hipMemsetAsync(INV, 0, invBytes, stream);
    rgcn_count_edges<<<(nEdges + 255) / 256, 256, 0, stream>>>(dst, et, INV, nEdges);
    rgcn_invert_counts<<<(nNodes * N_REL_C + 255) / 256, 256, 0, stream>>>(INV, nNodes * N_REL_C);

    // --- Layer 1 ---
    rgcn_wmma_gemm<<<dim3(nTilesM, N_REL_C + 1), 128, 0, stream>>>(x, w1, root1, XR, XROOT, nNodes);
    hipMemsetAsync(AGG, 0, nodeBytes, stream);
    {
        long long work = (long long)nEdges * 16;
        rgcn_edge_agg<<<(unsigned)((work + 255) / 256), 256, 0, stream>>>(XR, INV, src, dst, et, AGG, nEdges);
    }
    rgcn_combine_relu<<<(nodeElems + 255) / 256, 256, 0, stream>>>(AGG, XROOT, b1, H1, nodeElems);

    // --- Layer 2 ---
    rgcn_wmma_gemm<<<dim3(nTilesM, N_REL_C + 1), 128, 0, stream>>>(H1, w2, root2, XR, XROOT, nNodes);
    hipMemsetAsync(AGG, 0, nodeBytes, stream);
    {
        long long work = (long long)nEdges * 16;
        rgcn_edge_agg<<<(unsigned)((work + 255) / 256), 256, 0, stream>>>(XR, INV, src, dst, et, AGG, nEdges);
    }
    rgcn_combine_relu<<<(nodeElems + 255) / 256, 256, 0, stream>>>(AGG, XROOT, b2, H2, nodeElems);

    // --- Pool + classifier ---
    hipMemsetAsync(PSUM, 0, (N_GRAPHS_C * DIM_C + N_GRAPHS_C) * sizeof(float), stream);
    rgcn_pool<<<(nodeElems + 255) / 256, 256, 0, stream>>>(H2, batch, PSUM, PCNT, nodeElems);
    rgcn_final_linear<<<1, 128, 0, stream>>>(PSUM, PCNT, lin_w, lin_b, out);
}